// InterpolativeUpsampler_47485158424927
// MI455X (gfx1250) — compile-verified
//
#include <hip/hip_runtime.h>
#include <cstdint>

// ---------------------------------------------------------------------------
// InterpolativeUpsampler: nearest 2x upsample + 4x4 [1,3,3,1] blur, fused.
//
// Collapsed to a separable polyphase filter directly on x:
//   even phase (o=2m):   0.5*x[m-1] + 0.5*x[m]
//   odd  phase (o=2m+1): 0.125*x[m-1] + 0.75*x[m] + 0.125*x[m+1]
// applied independently in y and x, zero padding outside [0,128).
// Output: (8,256,257,257) f32.
//
// Memory-bound (0.67 GB traffic, ~29 us floor @ 23.3 TB/s). CDNA5 path used:
// TDM tensor_load_to_lds DMA stages each input tile into LDS (TENSORcnt),
// with hardware pad_enable inserting 4 dwords after every 128-dword row so
// rows land at a 132-float stride. Each thread owns one output column and
// runs a rolling vertical scan: horizontal filter evaluated once per input
// row, both output phases emitted from a 3-register window.
// ---------------------------------------------------------------------------

typedef unsigned int u32;
typedef __attribute__((ext_vector_type(4))) u32 v4u;
typedef __attribute__((ext_vector_type(8))) int v8i;
typedef __attribute__((ext_vector_type(4))) int v4i;

#define LDS_ROW   132           // 128 interior + 2 halo cols + 2 pad (dwords)
#define TILE_ROWS 34            // 32 interior input rows + 1 halo each side

#if defined(__AMDGCN__) && defined(__has_builtin)
#if __has_builtin(__builtin_amdgcn_tensor_load_to_lds) && \
    __has_builtin(__builtin_amdgcn_s_wait_tensorcnt)
#define UPS_HAVE_TDM 1
#endif
#endif

__global__ __launch_bounds__(256) void
upsample_blur_kernel(const float* __restrict__ x, float* __restrict__ out) {
  __shared__ float T[TILE_ROWS * LDS_ROW];

  const int t     = threadIdx.x;
  const int tileY = blockIdx.x;   // 0..3   (32 input rows each)
  const int bc    = blockIdx.y;   // 0..2047 (= 8*256 planes)
  const int M0    = tileY * 32;   // first interior input row of this tile

  // LDS row r <-> global input row (M0-1)+r ; LDS col c <-> input col c-1.
  // Zero the halo columns (input col -1 and 128) for every row.
  if (t < TILE_ROWS) {
    T[t * LDS_ROW + 0]   = 0.0f;
    T[t * LDS_ROW + 129] = 0.0f;
  }
  // Zero rows that fall outside the plane (zero padding of the conv).
  if (M0 == 0) {
    for (int c = t; c < 130; c += 256) T[0 * LDS_ROW + c] = 0.0f;
  }
  if (M0 == 96) {
    for (int c = t; c < 130; c += 256) T[33 * LDS_ROW + c] = 0.0f;
  }

  const int rowFirst = (M0 == 0) ? 0 : (M0 - 1);
  const int rowLast  = (M0 + 32 > 127) ? 127 : (M0 + 32);
  const int nrows    = rowLast - rowFirst + 1;       // 33 or 34
  const int ldsRow0  = rowFirst - (M0 - 1);          // 1 if M0==0 else 0
  const float* xPlane = x + (uint64_t)bc * (128u * 128u);

#if defined(UPS_HAVE_TDM)
  if (t == 0) {
    // Build Tensor DMA descriptor (D#) per CDNA5 ISA ch.8.
    uint64_t gaddr = (uint64_t)(uintptr_t)(xPlane + (uint64_t)rowFirst * 128u);
    u32 laddr = (u32)(uintptr_t)(&T[ldsRow0 * LDS_ROW + 1]);  // low 32 = LDS offset

    // Group 0: count=1 (valid), lds_addr, global_addr, type=2 ("image").
    v4u g0 = { 1u,
               laddr,
               (u32)gaddr,
               (u32)((gaddr >> 32) & 0x01FFFFFFu) | (2u << 30) };

    // Group 1: data_size=2 (4B), pad_enable=1, pad_interval=6 (128 dwords),
    // pad_amount=3 (4 dwords) -> LDS row stride 132 dwords.
    v8i g1;
    g1[0] = (int)((2u << 16) | (1u << 20) | (6u << 22) | (3u << 25));
    g1[1] = (int)(128u << 16);            // tensor_dim0 = 128 (low 16 @ bits 63:48)
    g1[2] = (int)((u32)nrows << 16);      // tensor_dim0 hi=0 | tensor_dim1 = nrows
    g1[3] = (int)(128u << 16);            // tensor_dim1 hi=0 | tile_dim0 = 128
    g1[4] = (int)(u32)nrows;              // tile_dim1 = nrows, tile_dim2 = 0
    g1[5] = 128;                          // tensor_dim0_stride = 128 elements
    g1[6] = 0;                            // stride0 hi | tensor_dim1_stride lo
    g1[7] = 0;
    v4i g2 = { 0, 0, 0, 0 };              // 2D tile: dims 2+ unused
    v4i g3 = { 0, 0, 0, 0 };
    v8i g4 = { 0, 0, 0, 0, 0, 0, 0, 0 };  // extra operand group (clang-23 6-arg form)

    __builtin_amdgcn_tensor_load_to_lds(g0, g1, g2, g3, g4, 0);
    __builtin_amdgcn_s_wait_tensorcnt(0);  // TDM data committed to LDS
  }
#else
  // Fallback (host pass / toolchains without the TDM builtin): cooperative load.
  for (int idx = t; idx < nrows * 128; idx += 256) {
    int r = idx >> 7, c = idx & 127;
    T[(ldsRow0 + r) * LDS_ROW + 1 + c] = xPlane[(uint64_t)(rowFirst + r) * 128u + c];
  }
#endif
  __syncthreads();

  // ---- compute: each thread owns output column(s) ox, rolling row scan ----
  const int oyBase = 2 * M0;
  float* outPlane = out + (uint64_t)bc * 66049u;     // 257*257

  for (int ox = t; ox < 257; ox += 256) {
    const int nx = ox >> 1;
    const int px = ox & 1;
    const int lc = nx + 1;                           // 1..129

    // Horizontal polyphase filter of LDS row r at this thread's column.
    auto hf = [&](int r) -> float {
      const float* rw = &T[r * LDS_ROW + lc];
      return (px == 0) ? 0.5f * (rw[-1] + rw[0])
                       : (0.125f * rw[-1] + 0.75f * rw[0] + 0.125f * rw[1]);
    };

    float Hm1 = hf(0);                               // H[M0-1] (zero row at edges)
    float Hc  = hf(1);                               // H[M0]
    float* ocol = outPlane + ox;

#pragma unroll 4
    for (int j = 0; j < 32; ++j) {
      const uint64_t oy = (uint64_t)(oyBase + 2 * j);
      // even output row 2m:   0.5*(H[m-1] + H[m])
      ocol[oy * 257u] = 0.5f * (Hm1 + Hc);
      const float Hn = hf(j + 2);                    // H[m+1]
      // odd output row 2m+1:  0.125*H[m-1] + 0.75*H[m] + 0.125*H[m+1]
      ocol[(oy + 1) * 257u] = 0.125f * Hm1 + 0.75f * Hc + 0.125f * Hn;
      Hm1 = Hc; Hc = Hn;
    }
    // Final output row 256 (last tile only): 0.5*(H[127] + H[128]), H[128]=0.
    if (tileY == 3) {
      ocol[256u * 257u] = 0.5f * (Hm1 + Hc);
    }
  }
}

extern "C" void kernel_launch(void* const* d_in, const int* in_sizes, int n_in,
                              void* d_out, int out_size, void* d_ws, size_t ws_size,
                              hipStream_t stream) {
  const float* x = (const float*)d_in[0];        // (8,256,128,128) f32
  // d_in[1] is the 4x4 kernel; its values are fixed by FILT and baked in above.
  float* out = (float*)d_out;                    // (8,256,257,257) f32

  const int planes = in_sizes[0] / (128 * 128);  // 2048
  dim3 grid(4, planes);
  upsample_blur_kernel<<<grid, 256, 0, stream>>>(x, out);
}